// AtomPositionGather_24859270709375
// MI455X (gfx1250) — compile-verified
//
#include <hip/hip_runtime.h>
#include <hip/hip_bf16.h>

typedef __attribute__((ext_vector_type(2))) float v2f;
typedef __attribute__((ext_vector_type(8))) float v8f;

#define ID_N_T  0
#define ID_CA_T 1
#define ID_C_T  2
#define ID_CB_T 4
#define EPS_F   1e-6f

__device__ __forceinline__ int imin_(int a, int b) { return a < b ? a : b; }

// ---------------- group boundary kernels (groups of 16 residues) -------------
__global__ __launch_bounds__(256) void fill_bounds_k(int* gstart, int* gend, int NG) {
  int i = blockIdx.x * blockDim.x + threadIdx.x;
  if (i < NG) { gstart[i] = 0; gend[i] = 0; }
}

__global__ __launch_bounds__(256) void find_bounds_k(const int* __restrict__ seg,
                                                     int* gstart, int* gend, int N) {
  int i = blockIdx.x * blockDim.x + threadIdx.x;
  if (i >= N) return;
  int g = seg[i] >> 4;
  if (i == 0     || (seg[i - 1] >> 4) != g) gstart[g] = i;
  if (i == N - 1 || (seg[i + 1] >> 4) != g) gend[g]   = i + 1;
}

// ---------------- atom2residue passthrough (int -> float) -------------------
__global__ __launch_bounds__(256) void copy_seg_k(const int* __restrict__ seg,
                                                  float* __restrict__ out, int N) {
  int i = blockIdx.x * blockDim.x + threadIdx.x;
  if (i < N) out[i] = (float)seg[i];
}

// ---------------- main kernel: one wave32 per 16-residue group ---------------
__global__ __launch_bounds__(256) void residue_gather_k(
    const float* __restrict__ feats,   // [N,128]
    const float* __restrict__ pos,     // [N,3]
    const int*   __restrict__ type,    // [N]
    const int*   __restrict__ seg,     // [N] sorted
    const int*   __restrict__ gstart,
    const int*   __restrict__ gend,
    float* __restrict__ out, int N, int R, int NG)
{
  const int wave = (int)((blockIdx.x * blockDim.x + threadIdx.x) >> 5);
  const int lane = (int)(threadIdx.x & 31);
  if (wave >= NG) return;

  const int r0   = wave << 4;
  const int colN = lane & 15;      // N-column within 16-wide tile / residue row id
  const int half = lane >> 4;      // 0: K=0,1  1: K=2,3 (A & B fragment halves)
  const int r    = r0 + colN;      // residue this lane bookkeeps (dup across halves)
  const int kA   = half << 1;

  const int gs = __builtin_amdgcn_readfirstlane(gstart[wave]);
  const int ge = __builtin_amdgcn_readfirstlane(gend[wave]);

  const v8f vzero = {0.f, 0.f, 0.f, 0.f, 0.f, 0.f, 0.f, 0.f};
  v8f c[8];
#pragma unroll
  for (int t = 0; t < 8; ++t) c[t] = vzero;

  int cntTotal = 0, cntCA = 0;
  int hasN = 0, hasCA = 0, hasC = 0;
  int idxCA = -1, idxCB = -1;

  for (int base = (gs & ~3); base < ge; base += 4) {
    int t4[4], s4[4];
    if (base + 3 < N) {
      const int4 tt = *reinterpret_cast<const int4*>(type + base);  // 16B aligned (base%4==0)
      const int4 ss = *reinterpret_cast<const int4*>(seg + base);
      t4[0] = tt.x; t4[1] = tt.y; t4[2] = tt.z; t4[3] = tt.w;
      s4[0] = ss.x; s4[1] = ss.y; s4[2] = ss.z; s4[3] = ss.w;
    } else {
#pragma unroll
      for (int k = 0; k < 4; ++k) {
        int idx = base + k;
        if (idx < N) { t4[k] = type[idx]; s4[k] = seg[idx]; }
        else         { t4[k] = -1;        s4[k] = -1; }
      }
    }

    bool ca4[4];
#pragma unroll
    for (int k = 0; k < 4; ++k) ca4[k] = ((base + k) < ge) && (t4[k] == ID_CA_T);

    // per-residue bookkeeping (each lane scans all 4 atoms of the chunk)
#pragma unroll
    for (int k = 0; k < 4; ++k) {
      if (s4[k] == r) {
        cntTotal++;
        if (t4[k] == ID_N_T)  hasN = 1;
        if (t4[k] == ID_CA_T) { hasCA = 1; cntCA++; idxCA = base + k; }
        if (t4[k] == ID_C_T)  hasC = 1;
        if (t4[k] == ID_CB_T) idxCB = base + k;
      }
    }

    int any = (int)(ca4[0] | ca4[1] | ca4[2] | ca4[3]);
    any = __builtin_amdgcn_readfirstlane(any);
    if (any) {
      // A fragment (16x4 f32): v0 <-> K = kA, v1 <-> K = kA+1, M = lane&15
      v2f A;
      A.x = (ca4[kA]     && s4[kA]     == r) ? 1.0f : 0.0f;
      A.y = (ca4[kA + 1] && s4[kA + 1] == r) ? 1.0f : 0.0f;

      const bool l0 = ca4[kA], l1 = ca4[kA + 1];
      const float* row0 = feats + (size_t)imin_(base + kA,     N - 1) * 128;
      const float* row1 = feats + (size_t)imin_(base + kA + 1, N - 1) * 128;

#pragma unroll
      for (int t = 0; t < 8; ++t) {
        // B fragment (4x16 f32): v0 <-> K = kA, v1 <-> K = kA+1, N = lane&15
        v2f B;
        B.x = l0 ? row0[t * 16 + colN] : 0.0f;
        B.y = l1 ? row1[t * 16 + colN] : 0.0f;
        c[t] = __builtin_amdgcn_wmma_f32_16x16x4_f32(
            false, A, false, B, (short)0, c[t], false, false);
      }
    }
  }

  // ---- residue_features writeout: D layout VGPR v -> row v (lanes 0-15) / v+8
  const float cntCAf = (float)cntCA;
#pragma unroll
  for (int t = 0; t < 8; ++t) {
#pragma unroll
    for (int v = 0; v < 8; ++v) {
      const int row = v + (half << 3);
      const int rr  = r0 + row;
      const float cr = __shfl(cntCAf, row, 32);  // lane `row` holds count of residue r0+row
      if (rr < R) {
        const float val = (cr > 0.f) ? (c[t][v] / cr) : 0.f;
        out[(size_t)rr * 128 + t * 16 + colN] = val;
      }
    }
  }

  // ---- positions / frames / mask (lanes 0-15, one residue each) -------------
  const size_t offCA   = (size_t)R * 128;
  const size_t offCB   = offCA + (size_t)R * 3;
  const size_t offFr   = offCB + (size_t)R * 3;
  const size_t offMask = offFr + (size_t)R * 9 + (size_t)N;

  if (half == 0 && r < R) {
    float cax = 0.f, cay = 0.f, caz = 0.f;
    if (idxCA >= 0) {
      const float* p = pos + (size_t)idxCA * 3;
      cax = p[0]; cay = p[1]; caz = p[2];
    }
    float cbx = 0.f, cby = 0.f, cbz = 0.f;
    if (idxCB >= 0) {
      const float* p = pos + (size_t)idxCB * 3;
      cbx = p[0]; cby = p[1]; cbz = p[2];
    }
    const bool no_cb = (fabsf(cbx) + fabsf(cby) + fabsf(cbz)) < EPS_F;
    if (no_cb) { cbx = cax; cby = cay; cbz = caz; }

    out[offCA + (size_t)r * 3 + 0] = cax;
    out[offCA + (size_t)r * 3 + 1] = cay;
    out[offCA + (size_t)r * 3 + 2] = caz;
    out[offCB + (size_t)r * 3 + 0] = cbx;
    out[offCB + (size_t)r * 3 + 1] = cby;
    out[offCB + (size_t)r * 3 + 2] = cbz;

    // frames
    const float e1x = cbx - cax, e1y = cby - cay, e1z = cbz - caz;
    const float n1 = sqrtf(e1x * e1x + e1y * e1y + e1z * e1z);
    const bool valid1 = n1 > EPS_F;
    const float inv1 = 1.f / fmaxf(n1, 1e-12f);
    const float ux = e1x * inv1, uy = e1y * inv1, uz = e1z * inv1;
    // cross(u, z) = (uy, -ux, 0)
    const float n2a = sqrtf(uy * uy + ux * ux);
    const bool use_y = n2a < EPS_F;
    // cross(u, y) = (-uz, 0, ux)
    const float e2x = use_y ? -uz : uy;
    const float e2y = use_y ? 0.f : -ux;
    const float e2z = use_y ? ux : 0.f;
    const float n2 = sqrtf(e2x * e2x + e2y * e2y + e2z * e2z);
    const bool valid2 = n2 > EPS_F;
    const float inv2 = 1.f / fmaxf(n2, 1e-12f);
    const float vx = e2x * inv2, vy = e2y * inv2, vz = e2z * inv2;
    const float wx = uy * vz - uz * vy;
    const float wy = uz * vx - ux * vz;
    const float wz = ux * vy - uy * vx;
    const bool valid = valid1 && valid2 && (r < (R - 1));

    float* fr = out + offFr + (size_t)r * 9;
    if (valid) {
      fr[0] = ux; fr[1] = vx; fr[2] = wx;
      fr[3] = uy; fr[4] = vy; fr[5] = wy;
      fr[6] = uz; fr[7] = vz; fr[8] = wz;
    } else {
      fr[0] = 1.f; fr[1] = 0.f; fr[2] = 0.f;
      fr[3] = 0.f; fr[4] = 1.f; fr[5] = 0.f;
      fr[6] = 0.f; fr[7] = 0.f; fr[8] = 1.f;
    }

    const bool mok = (cntTotal >= 3) && hasN && hasCA && hasC;
    out[offMask + r] = mok ? 1.f : 0.f;
  }
}

extern "C" void kernel_launch(void* const* d_in, const int* in_sizes, int n_in,
                              void* d_out, int out_size, void* d_ws, size_t ws_size,
                              hipStream_t stream) {
  const float* feats = (const float*)d_in[0];
  const float* pos   = (const float*)d_in[1];
  const int*   type  = (const int*)d_in[2];
  const int*   seg   = (const int*)d_in[3];
  // d_in[4] is num_residues on device; recover R from sizes instead:
  const int N = in_sizes[2];
  const int R = (int)(((long long)out_size - (long long)N) / 144LL);
  const int NG = (R + 15) / 16;

  int* gstart = (int*)d_ws;
  int* gend   = gstart + NG;
  float* out  = (float*)d_out;

  fill_bounds_k<<<(NG + 255) / 256, 256, 0, stream>>>(gstart, gend, NG);
  find_bounds_k<<<(N + 255) / 256, 256, 0, stream>>>(seg, gstart, gend, N);

  const int totalThreads = NG * 32;
  residue_gather_k<<<(totalThreads + 255) / 256, 256, 0, stream>>>(
      feats, pos, type, seg, gstart, gend, out, N, R, NG);

  copy_seg_k<<<(N + 255) / 256, 256, 0, stream>>>(seg, out + (size_t)R * 143, N);
}